// EnhancedFrequencyLogGaborConv2d_12841952215531
// MI455X (gfx1250) — compile-verified
//
#include <hip/hip_runtime.h>
#include <hip/hip_bf16.h>
#include <math.h>

// ---------------------------------------------------------------------------
// EnhancedFrequencyLogGaborConv2d for MI455X (gfx1250, wave32, WMMA).
// fft2 / ifft2 are done as exact fp32 DFT matrix products on the WMMA pipe:
//   Y = F X F^T  (F symmetric 64x64 DFT matrix), via V_WMMA_F32_16X16X4_F32.
// Attention MLP (64->64 relu, 64->3 softmax) also runs on f32 WMMA.
// ---------------------------------------------------------------------------

typedef __attribute__((ext_vector_type(2))) float v2f;
typedef __attribute__((ext_vector_type(8))) float v8f;

// One 16x16 output tile of D += P(64xK=64) * Q(64x N), K fixed at 64.
// P row-major with leading dim ldp, Q row-major with leading dim ldq.
// A-operand layout (16x4 f32): lanes 0-15 hold K={0,1}, lanes 16-31 K={2,3}.
// B-operand layout (4x16 f32): mirrored (VGPR0: K=0/K=2, VGPR1: K=1/K=3).
template <bool NEGP>
__device__ inline v8f wmma_mac64(const float* P, int ldp,
                                 const float* Q, int ldq,
                                 int mBase, int nBase, v8f acc) {
  const int lane = threadIdx.x & 31;
  const int half = lane >> 4;
  const int l    = lane & 15;
  const int arow = (mBase + l) * ldp;
  const int bcol = nBase + l;
#pragma unroll
  for (int k = 0; k < 64; k += 4) {
    const int kk = k + 2 * half;
    v2f a, b;
    a.x = P[arow + kk];
    a.y = P[arow + kk + 1];
    if (NEGP) { a.x = -a.x; a.y = -a.y; }
    b.x = Q[kk * ldq + bcol];
    b.y = Q[(kk + 1) * ldq + bcol];
    acc = __builtin_amdgcn_wmma_f32_16x16x4_f32(false, a, false, b,
                                                (short)0, acc, false, false);
  }
  return acc;
}

// Store a 16x16 f32 C/D tile: VGPR r holds rows (r, r+8) split across lane halves.
__device__ inline void store_tile(float* D, int ldd, int mBase, int nBase, v8f acc) {
  const int lane = threadIdx.x & 31;
  const int half = lane >> 4;
  const int l    = lane & 15;
#pragma unroll
  for (int r = 0; r < 8; ++r)
    D[(mBase + r + 8 * half) * ldd + nBase + l] = acc[r];
}

// ---------------------------------------------------------------------------
// Kernel 0: DFT matrix tables.  Fr,Fi = forward (exp(-i)), Cr,Ci = inverse/64.
// ---------------------------------------------------------------------------
__global__ void tables_kernel(float* __restrict__ tabs) {
  int t = blockIdx.x * blockDim.x + threadIdx.x;
  if (t >= 4096) return;
  int k = t >> 6, n = t & 63;
  float th = (6.28318530717958647692f / 64.0f) * (float)((k * n) & 63);
  float c = cosf(th), s = sinf(th);
  tabs[t]         = c;                   // Fr
  tabs[4096 + t]  = -s;                  // Fi
  tabs[8192 + t]  = c * (1.0f / 64.0f);  // Cr
  tabs[12288 + t] = s * (1.0f / 64.0f);  // Ci
}

// ---------------------------------------------------------------------------
// Kernel 1: forward DFT per (b,i) image.  Yr -> d_out, Yi -> ws.
//   A = Fr*X, B = Fi*X;  Yr = A*Fr - B*Fi;  Yi = A*Fi + B*Fr  (F symmetric).
// ---------------------------------------------------------------------------
__global__ __launch_bounds__(256) void fwd_dft_kernel(
    const float* __restrict__ x, const float* __restrict__ tabs,
    float* __restrict__ yr, float* __restrict__ yi) {
  __shared__ float sX[4096];   // X, then reused as A
  __shared__ float sB[4096];
  __shared__ float sFr[4096];
  __shared__ float sFi[4096];
  const int img = blockIdx.x;
  const int tid = threadIdx.x;
  const float* Xg = x + (size_t)img * 4096;
  for (int t = tid; t < 4096; t += 256) {
    sX[t]  = Xg[t];
    sFr[t] = tabs[t];
    sFi[t] = tabs[4096 + t];
  }
  __syncthreads();
  const int wave = tid >> 5;
  const int t0 = wave, t1 = wave + 8;
  const int m0 = (t0 >> 2) * 16, n0 = (t0 & 3) * 16;
  const int m1 = (t1 >> 2) * 16, n1 = (t1 & 3) * 16;

  // stage 1
  v8f a0 = {}, a1 = {}, b0 = {}, b1v = {};
  a0  = wmma_mac64<false>(sFr, 64, sX, 64, m0, n0, a0);
  a1  = wmma_mac64<false>(sFr, 64, sX, 64, m1, n1, a1);
  b0  = wmma_mac64<false>(sFi, 64, sX, 64, m0, n0, b0);
  b1v = wmma_mac64<false>(sFi, 64, sX, 64, m1, n1, b1v);
  __syncthreads();
  store_tile(sX, 64, m0, n0, a0);    // sX := A
  store_tile(sX, 64, m1, n1, a1);
  store_tile(sB, 64, m0, n0, b0);    // sB := B
  store_tile(sB, 64, m1, n1, b1v);
  __syncthreads();

  // stage 2
  v8f yr0 = {}, yr1 = {}, yi0 = {}, yi1 = {};
  yr0 = wmma_mac64<false>(sX, 64, sFr, 64, m0, n0, yr0);
  yr0 = wmma_mac64<true >(sB, 64, sFi, 64, m0, n0, yr0);
  yr1 = wmma_mac64<false>(sX, 64, sFr, 64, m1, n1, yr1);
  yr1 = wmma_mac64<true >(sB, 64, sFi, 64, m1, n1, yr1);
  yi0 = wmma_mac64<false>(sX, 64, sFi, 64, m0, n0, yi0);
  yi0 = wmma_mac64<false>(sB, 64, sFr, 64, m0, n0, yi0);
  yi1 = wmma_mac64<false>(sX, 64, sFi, 64, m1, n1, yi1);
  yi1 = wmma_mac64<false>(sB, 64, sFr, 64, m1, n1, yi1);

  float* yrg = yr + (size_t)img * 4096;
  float* yig = yi + (size_t)img * 4096;
  store_tile(yrg, 64, m0, n0, yr0);
  store_tile(yrg, 64, m1, n1, yr1);
  store_tile(yig, 64, m0, n0, yi0);
  store_tile(yig, 64, m1, n1, yi1);
}

// ---------------------------------------------------------------------------
// Kernel 2: attention weights aw[b][s][h*64+w] in the *shifted* frame.
// One block = one (b, chunk of 128 pixels).  h1 = relu(W1 |xs| + b1),
// logits = W2 h1 + b2, softmax over S=3.  Both GEMMs on f32 WMMA.
// ---------------------------------------------------------------------------
__global__ __launch_bounds__(256) void attn_kernel(
    const float* __restrict__ yr, const float* __restrict__ yi,
    const float* __restrict__ w1, const float* __restrict__ b1,
    const float* __restrict__ w2, const float* __restrict__ b2,
    float* __restrict__ aw) {
  __shared__ float sV[64 * 128];   // |xs| (channels x pixels), reused for H1
  __shared__ float sW1[4096];
  __shared__ float sW2[16 * 64];   // W2 zero-padded to 16 rows
  const int b = blockIdx.y;
  const int p0 = blockIdx.x * 128;
  const int tid = threadIdx.x;
  for (int t = tid; t < 4096; t += 256) sW1[t] = w1[t];
  for (int t = tid; t < 1024; t += 256) sW2[t] = (t < 192) ? w2[t] : 0.0f;
  const int bg = (b + 32) & 63;
  for (int t = tid; t < 64 * 128; t += 256) {
    int c = t >> 7, q = t & 127;
    int p = p0 + q, h = p >> 6, w = p & 63;
    int cg = (c + 32) & 63, hg = (h + 32) & 63, wg = (w + 32) & 63;
    size_t idx = (((size_t)(bg * 64 + cg)) * 64 + hg) * 64 + wg;
    float re = yr[idx], im = yi[idx];
    sV[c * 128 + q] = sqrtf(re * re + im * im);
  }
  __syncthreads();
  const int wave = tid >> 5;

  // H1 tiles: wave owns column strip nt=wave, mt = 0..3
  v8f hacc[4];
#pragma unroll
  for (int mt = 0; mt < 4; ++mt) {
    v8f acc = {};
    hacc[mt] = wmma_mac64<false>(sW1, 64, sV, 128, mt * 16, wave * 16, acc);
  }
  __syncthreads();
  {
    const int lane = tid & 31, half = lane >> 4, l = lane & 15;
#pragma unroll
    for (int mt = 0; mt < 4; ++mt)
#pragma unroll
      for (int r = 0; r < 8; ++r) {
        int row = mt * 16 + r + 8 * half;
        int col = wave * 16 + l;
        float v = hacc[mt][r] + b1[row];
        sV[row * 128 + col] = v > 0.0f ? v : 0.0f;   // H1
      }
  }
  __syncthreads();

  // logits: 16x128 (rows 0..2 valid), wave owns tile nt=wave
  v8f lg = {};
  lg = wmma_mac64<false>(sW2, 64, sV, 128, 0, wave * 16, lg);

  const int lane = tid & 31;
  if (lane < 16) {
    int p = p0 + wave * 16 + lane;
    float l0 = lg[0] + b2[0];
    float l1 = lg[1] + b2[1];
    float l2 = lg[2] + b2[2];
    float m = fmaxf(l0, fmaxf(l1, l2));
    float e0 = expf(l0 - m), e1 = expf(l1 - m), e2 = expf(l2 - m);
    float inv = 1.0f / (e0 + e1 + e2);
    aw[(size_t)(b * 3 + 0) * 4096 + p] = e0 * inv;
    aw[(size_t)(b * 3 + 1) * 4096 + p] = e1 * inv;
    aw[(size_t)(b * 3 + 2) * 4096 + p] = e2 * inv;
  }
}

// ---------------------------------------------------------------------------
// Kernel 3: pre-shifted Gabor gain Gs[i][h][w] = sum_s F[s,i',h',w']*aw[i',s,h',w']
// (primes = +32 mod 64; Gs multiplies the UNshifted spectrum Y).
// ---------------------------------------------------------------------------
__global__ void gabor_kernel(const float* __restrict__ f0,
                             const float* __restrict__ theta,
                             const float* __restrict__ sigma,
                             const float* __restrict__ theta0,
                             const float* __restrict__ aw,
                             float* __restrict__ gs) {
  int t = blockIdx.x * blockDim.x + threadIdx.x;
  if (t >= 64 * 64 * 64) return;
  int w = t & 63, h = (t >> 6) & 63, i = t >> 12;
  int ip = (i + 32) & 63, hp = (h + 32) & 63, wp = (w + 32) & 63;
  float yy = -1.0f + 2.0f * (float)hp / 63.0f;
  float xx = -1.0f + 2.0f * (float)wp / 63.0f;
  float r = sqrtf(xx * xx + yy * yy + 1e-6f);
  float phi = atan2f(yy, xx);
  float lr = logf(r);
  float g = 0.0f;
#pragma unroll
  for (int s = 0; s < 3; ++s) {
    float d1 = lr - logf(f0[s * 64 + ip]);
    float ls = logf(sigma[s * 64 + ip]);
    float lgab = expf(-(d1 * d1) / (2.0f * ls * ls));
    float d2 = phi - theta[s * 64 + ip];
    float t0v = theta0[s * 64 + ip];
    float angf = expf(-(d2 * d2) / (2.0f * t0v * t0v));
    g += lgab * angf * aw[(size_t)(ip * 3 + s) * 4096 + hp * 64 + wp];
  }
  gs[t] = g;
}

// ---------------------------------------------------------------------------
// Kernel 4: spatial 3x3 conv, O=1, zero pad.  xsp[b][h][w].
// ---------------------------------------------------------------------------
__global__ void conv_kernel(const float* __restrict__ x,
                            const float* __restrict__ cw,
                            float* __restrict__ xsp) {
  int t = blockIdx.x * blockDim.x + threadIdx.x;
  if (t >= 64 * 64 * 64) return;
  int w = t & 63, h = (t >> 6) & 63, b = t >> 12;
  float acc = 0.0f;
  const float* xb = x + (size_t)b * 64 * 4096;
  for (int i = 0; i < 64; ++i) {
    const float* xi = xb + (size_t)i * 4096;
    const float* ki = cw + i * 9;
#pragma unroll
    for (int kh = 0; kh < 3; ++kh) {
      int hh = h + kh - 1;
      if (hh < 0 || hh > 63) continue;
#pragma unroll
      for (int kw = 0; kw < 3; ++kw) {
        int ww = w + kw - 1;
        if (ww < 0 || ww > 63) continue;
        acc += xi[hh * 64 + ww] * ki[kh * 3 + kw];
      }
    }
  }
  xsp[t] = acc;
}

// ---------------------------------------------------------------------------
// Kernel 5: Z = Y * Gs * mask;  x_f = Re(C Z C^T);  out = mix*x_f + (1-mix)*xsp.
// Reads Yr from d_out (into LDS, barriered) then overwrites d_out with result.
//   U = Cr*Zr - Ci*Zi ; V = Cr*Zi + Ci*Zr ; R = U*Cr - V*Ci  (C symmetric)
// ---------------------------------------------------------------------------
__global__ __launch_bounds__(256) void idft_kernel(
    float* yrout,                              // Yr in, final out (aliased)
    const float* __restrict__ yi, const float* __restrict__ ctabs,
    const float* __restrict__ gs, const float* __restrict__ fbs,
    const float* __restrict__ mixp, const float* __restrict__ xsp) {
  __shared__ float sZr[4096];   // Zr, then U
  __shared__ float sZi[4096];   // Zi, then V
  __shared__ float sCr[4096];
  __shared__ float sCi[4096];
  const int img = blockIdx.x;
  const int b = img >> 6, i = img & 63;
  const int tid = threadIdx.x;
  const int j  = (b + 32) & 63;
  const int ip = (i + 32) & 63;
  const int si = (int)floorf((fbs[2 * j] + 1.0f) * 32.0f);
  const int ei = (int)floorf((fbs[2 * j + 1] + 1.0f) * 32.0f);
  const bool chanOK = (ip >= si) && (ip < ei);
  const float* Zr_g = yrout + (size_t)img * 4096;
  const float* Zi_g = yi + (size_t)img * 4096;
  const float* Gsi  = gs + (size_t)i * 4096;
  for (int t = tid; t < 4096; t += 256) {
    int h = t >> 6;
    int hp = (h + 32) & 63;
    float m = (chanOK && hp >= si && hp < ei) ? 1.0f : 0.0f;
    float gm = Gsi[t] * m;
    sZr[t] = Zr_g[t] * gm;
    sZi[t] = Zi_g[t] * gm;
    sCr[t] = ctabs[t];
    sCi[t] = ctabs[4096 + t];
  }
  __syncthreads();
  const int wave = tid >> 5;
  const int t0 = wave, t1 = wave + 8;
  const int m0 = (t0 >> 2) * 16, n0 = (t0 & 3) * 16;
  const int m1 = (t1 >> 2) * 16, n1 = (t1 & 3) * 16;

  v8f u0 = {}, u1 = {}, v0 = {}, v1 = {};
  u0 = wmma_mac64<false>(sCr, 64, sZr, 64, m0, n0, u0);
  u0 = wmma_mac64<true >(sCi, 64, sZi, 64, m0, n0, u0);
  u1 = wmma_mac64<false>(sCr, 64, sZr, 64, m1, n1, u1);
  u1 = wmma_mac64<true >(sCi, 64, sZi, 64, m1, n1, u1);
  v0 = wmma_mac64<false>(sCr, 64, sZi, 64, m0, n0, v0);
  v0 = wmma_mac64<false>(sCi, 64, sZr, 64, m0, n0, v0);
  v1 = wmma_mac64<false>(sCr, 64, sZi, 64, m1, n1, v1);
  v1 = wmma_mac64<false>(sCi, 64, sZr, 64, m1, n1, v1);
  __syncthreads();
  store_tile(sZr, 64, m0, n0, u0);   // sZr := U
  store_tile(sZr, 64, m1, n1, u1);
  store_tile(sZi, 64, m0, n0, v0);   // sZi := V
  store_tile(sZi, 64, m1, n1, v1);
  __syncthreads();

  v8f r0 = {}, r1 = {};
  r0 = wmma_mac64<false>(sZr, 64, sCr, 64, m0, n0, r0);
  r0 = wmma_mac64<true >(sZi, 64, sCi, 64, m0, n0, r0);
  r1 = wmma_mac64<false>(sZr, 64, sCr, 64, m1, n1, r1);
  r1 = wmma_mac64<true >(sZi, 64, sCi, 64, m1, n1, r1);

  const float mix = mixp[0];
  const float omix = 1.0f - mix;
  const float* xs_b = xsp + (size_t)b * 4096;
  float* og = yrout + (size_t)img * 4096;
  const int lane = tid & 31, half = lane >> 4, l = lane & 15;
#pragma unroll
  for (int r = 0; r < 8; ++r) {
    int row0 = m0 + r + 8 * half, col0 = n0 + l;
    og[row0 * 64 + col0] = mix * r0[r] + omix * xs_b[row0 * 64 + col0];
    int row1 = m1 + r + 8 * half, col1 = n1 + l;
    og[row1 * 64 + col1] = mix * r1[r] + omix * xs_b[row1 * 64 + col1];
  }
}

// ---------------------------------------------------------------------------
// Host launcher.  ws layout (floats):
//   [0 .. 16777216)        Yi
//   [16777216 .. +16384)   DFT tables: Fr, Fi, Cr, Ci (4096 each)
//   [16793600 .. +786432)  aw [64][3][4096]
//   [17580032 .. +262144)  Gs [64][64][64]
//   [17842176 .. +262144)  xsp [64][64][64]
// Total ~72.4 MB.
// ---------------------------------------------------------------------------
extern "C" void kernel_launch(void* const* d_in, const int* in_sizes, int n_in,
                              void* d_out, int out_size, void* d_ws, size_t ws_size,
                              hipStream_t stream) {
  const float* x      = (const float*)d_in[0];
  const float* f0     = (const float*)d_in[1];
  const float* theta  = (const float*)d_in[2];
  const float* sigma  = (const float*)d_in[3];
  const float* theta0 = (const float*)d_in[4];
  const float* fbs    = (const float*)d_in[5];
  const float* mix    = (const float*)d_in[6];
  const float* w1     = (const float*)d_in[7];
  const float* b1     = (const float*)d_in[8];
  const float* w2     = (const float*)d_in[9];
  const float* b2     = (const float*)d_in[10];
  const float* convw  = (const float*)d_in[11];
  float* out = (float*)d_out;
  float* ws  = (float*)d_ws;

  float* yi   = ws;
  float* tabs = ws + 16777216;
  float* aw   = ws + 16793600;
  float* gs   = ws + 17580032;
  float* xsp  = ws + 17842176;

  tables_kernel<<<16, 256, 0, stream>>>(tabs);
  fwd_dft_kernel<<<4096, 256, 0, stream>>>(x, tabs, out, yi);
  attn_kernel<<<dim3(32, 64), 256, 0, stream>>>(out, yi, w1, b1, w2, b2, aw);
  gabor_kernel<<<1024, 256, 0, stream>>>(f0, theta, sigma, theta0, aw, gs);
  conv_kernel<<<1024, 256, 0, stream>>>(x, convw, xsp);
  idft_kernel<<<4096, 256, 0, stream>>>(out, yi, tabs + 8192, gs, fbs, mix, xsp);
}